// YOLONMSLayer_2241972928830
// MI455X (gfx1250) — compile-verified
//
#include <hip/hip_runtime.h>
#include <hip/hip_bf16.h>

#define NBOX 2048
#define NCLS 80
#define MAXB 100
#define SCORE_THR 0.6f
#define IOU_THR 0.5f
#define NEGF -1e30f

typedef float v2f __attribute__((ext_vector_type(2)));
typedef float v8f __attribute__((ext_vector_type(8)));

#if __has_builtin(__builtin_amdgcn_global_load_async_to_lds_b32)
#define HAVE_ASYNC_LDS 1
typedef __attribute__((address_space(1))) int gbl_i32_t;  // builtin wants __device__ int*
typedef __attribute__((address_space(3))) int lds_i32_t;  // and LDS int*
#else
#define HAVE_ASYNC_LDS 0
#endif

__device__ __forceinline__ void wait_asynccnt0() {
#if __has_builtin(__builtin_amdgcn_s_wait_asynccnt)
  __builtin_amdgcn_s_wait_asynccnt(0);
#else
  asm volatile("s_wait_asynccnt 0" ::: "memory");
#endif
}

// ---------------------------------------------------------------------------
// Kernel 1: pairwise IoU, one wave per 16x16 tile.
// union_sum[i][j] = area_i + area_j computed with V_WMMA_F32_16X16X4_F32:
//   A(16x4) = [area_row | 1 | 0 | 0],  B(4x16) = [1 ; area_col ; 0 ; 0]
//   => D[i][j] = area_i*1 + 1*area_j   (exact, only two nonzero products)
// ---------------------------------------------------------------------------
__global__ __launch_bounds__(256) void yolo_iou_wmma_kernel(
    const float* __restrict__ boxes, float* __restrict__ iou) {
  const int lane = threadIdx.x & 31;
  const int wave = blockIdx.x * 8 + (threadIdx.x >> 5);
  const int tm = (wave >> 7) << 4;   // tile row base (128 tiles per dim)
  const int tn = (wave & 127) << 4;  // tile col base
  const int l  = lane & 15;
  const int hi = lane >> 4;

  const float4* b4 = (const float4*)boxes;  // (y1, x1, y2, x2)

  float4 bj = b4[tn + l];
  float areaCol = fmaxf(bj.z - bj.x, 0.f) * fmaxf(bj.w - bj.y, 0.f);
  float4 br = b4[tm + l];
  float areaRow = fmaxf(br.z - br.x, 0.f) * fmaxf(br.w - br.y, 0.f);

  v2f A, B;
  A.x = hi ? 0.f : areaRow;  // A: K=0 col = area_row, K=2 col = 0
  A.y = hi ? 0.f : 1.0f;     //    K=1 col = ones    , K=3 col = 0
  B.x = hi ? 0.f : 1.0f;     // B: K=0 row = ones    , K=2 row = 0
  B.y = hi ? 0.f : areaCol;  //    K=1 row = area_col, K=3 row = 0
  v8f Cz = {};
  v8f D = __builtin_amdgcn_wmma_f32_16x16x4_f32(
      false, A, false, B, (short)0, Cz, false, false);

#pragma unroll
  for (int r = 0; r < 8; ++r) {
    int i = tm + r + (hi << 3);          // row index matching D layout
    float4 bi = b4[i];                   // broadcast load across 16 lanes
    float iy1 = fmaxf(bi.x, bj.x);
    float ix1 = fmaxf(bi.y, bj.y);
    float iy2 = fminf(bi.z, bj.z);
    float ix2 = fminf(bi.w, bj.w);
    float inter = fmaxf(iy2 - iy1, 0.f) * fmaxf(ix2 - ix1, 0.f);
    float uni = D[r] - inter;            // area_i + area_j - inter
    float v = (uni > 0.f) ? (inter / uni) : 0.f;
    iou[(size_t)i * NBOX + tn + l] = v;
  }
}

// ---------------------------------------------------------------------------
// Kernel 2: one workgroup (256 thr = 8 wave32) per class.
//   - ASYNC-copy the strided score column into LDS (global_load_async_to_lds_b32,
//     s_wait_asynccnt) -- single global pass, all later reads hit LDS
//   - bitonic sort ascending on (~orderable(score) << 32 | index)  == argsort(-s)
//   - sequential greedy suppression, IoU rows from the L2-resident matrix,
//     next kept row prefetched (global_prefetch_b8)
//   - first 100 survivors -> boxes/scores/class, rest zero / -1
// ---------------------------------------------------------------------------
__global__ __launch_bounds__(256) void yolo_nms_class_kernel(
    const float* __restrict__ boxes, const float* __restrict__ scores,
    const float* __restrict__ iou, float* __restrict__ outBoxes,
    float* __restrict__ outScores, int* __restrict__ outClasses) {
  __shared__ float sval[NBOX];              // 8 KB: score column, by box index
  __shared__ unsigned long long key[NBOX];  // 16 KB
  __shared__ unsigned char keep[NBOX];      // 2 KB
  __shared__ int sel[MAXB];
  __shared__ int selCount;
  __shared__ int nvalidLds;

  const int c = blockIdx.x;
  const int tid = threadIdx.x;

  if (tid == 0) nvalidLds = 0;

  // stage the score column into LDS
#if HAVE_ASYNC_LDS
  for (int i = tid; i < NBOX; i += 256) {
    __builtin_amdgcn_global_load_async_to_lds_b32(
        (gbl_i32_t*)(scores + i * NCLS + c), (lds_i32_t*)&sval[i],
        /*offset=*/0, /*cpol=*/0);
  }
  wait_asynccnt0();
#else
  for (int i = tid; i < NBOX; i += 256) sval[i] = scores[i * NCLS + c];
#endif
  __syncthreads();

  int localValid = 0;
  for (int i = tid; i < NBOX; i += 256) {
    float s = sval[i];
    bool valid = (s >= SCORE_THR);
    float kf = valid ? s : NEGF;
    unsigned u = __float_as_uint(kf);
    u = (u & 0x80000000u) ? ~u : (u | 0x80000000u);  // total order map
    key[i] = ((unsigned long long)(~u) << 32) | (unsigned)i;
    if (valid) localValid++;
  }
  atomicAdd(&nvalidLds, localValid);

  // bitonic sort, ascending composite key (desc score, asc index on ties)
  for (unsigned k = 2; k <= NBOX; k <<= 1) {
    for (unsigned j = k >> 1; j > 0; j >>= 1) {
      __syncthreads();
      for (unsigned i = tid; i < NBOX; i += 256) {
        unsigned ixj = i ^ j;
        if (ixj > i) {
          unsigned long long a = key[i], b = key[ixj];
          bool up = ((i & k) == 0);
          if ((a > b) == up) { key[i] = b; key[ixj] = a; }
        }
      }
    }
  }
  __syncthreads();

  // initial keep = valid (score >= thr), in sorted order
  for (int i = tid; i < NBOX; i += 256) {
    int idx = (int)(key[i] & 0xffffffffu);
    keep[i] = (sval[idx] >= SCORE_THR) ? 1 : 0;
  }
  __syncthreads();

  const int nv = nvalidLds;  // all valid entries sort to the front

  for (int i = 0; i < nv; ++i) {
    __syncthreads();  // make prior suppressions visible before reading keep[i]
    if (keep[i]) {
      int oi = (int)(key[i] & 0xffffffffu);
      const float* row = iou + (size_t)oi * NBOX;
      if (i + 1 < nv) {  // prefetch next candidate's 8 KB IoU row
        int on = (int)(key[i + 1] & 0xffffffffu);
        const float* rn = iou + (size_t)on * NBOX;
        if (tid < 64) __builtin_prefetch(rn + tid * 32, 0, 0);
      }
      for (int jj = i + 1 + tid; jj < nv; jj += 256) {
        if (keep[jj]) {
          int oj = (int)(key[jj] & 0xffffffffu);
          if (row[oj] > IOU_THR) keep[jj] = 0;
        }
      }
    }
  }
  __syncthreads();

  if (tid == 0) {
    int cnt = 0;
    for (int i = 0; i < nv && cnt < MAXB; ++i)
      if (keep[i]) sel[cnt++] = i;
    selCount = cnt;
  }
  __syncthreads();

  const int sc = selCount;
  for (int k = tid; k < MAXB; k += 256) {
    int base = c * MAXB + k;
    if (k < sc) {
      int idx = (int)(key[sel[k]] & 0xffffffffu);
      float4 b = ((const float4*)boxes)[idx];
      outBoxes[base * 4 + 0] = b.x;
      outBoxes[base * 4 + 1] = b.y;
      outBoxes[base * 4 + 2] = b.z;
      outBoxes[base * 4 + 3] = b.w;
      outScores[base] = sval[idx];
      outClasses[base] = c;
    } else {
      outBoxes[base * 4 + 0] = 0.f;
      outBoxes[base * 4 + 1] = 0.f;
      outBoxes[base * 4 + 2] = 0.f;
      outBoxes[base * 4 + 3] = 0.f;
      outScores[base] = 0.f;
      outClasses[base] = -1;
    }
  }
}

extern "C" void kernel_launch(void* const* d_in, const int* in_sizes, int n_in,
                              void* d_out, int out_size, void* d_ws, size_t ws_size,
                              hipStream_t stream) {
  const float* boxes = (const float*)d_in[0];      // (2048, 4)
  const float* box_scores = (const float*)d_in[1]; // (2048, 80)
  float* iou = (float*)d_ws;                       // 2048*2048*4 = 16 MB scratch

  float* out = (float*)d_out;
  float* outBoxes = out;                           // 32000 floats
  float* outScores = out + NCLS * MAXB * 4;        // 8000 floats
  int* outClasses = (int*)(out + NCLS * MAXB * 4 + NCLS * MAXB);  // 8000 int32

  yolo_iou_wmma_kernel<<<2048, 256, 0, stream>>>(boxes, iou);
  yolo_nms_class_kernel<<<NCLS, 256, 0, stream>>>(boxes, box_scores, iou,
                                                  outBoxes, outScores, outClasses);
}